// RouteOptimizationModel_13700945674596
// MI455X (gfx1250) — compile-verified
//
#include <hip/hip_runtime.h>
#include <hip/hip_bf16.h>

typedef __attribute__((ext_vector_type(2))) float v2f;
typedef __attribute__((ext_vector_type(8))) float v8f;

#define DIN 128   // D_IN == D_HID == 128

// Native CDNA5 f32 atomic add, non-returning (STOREcnt), device scope so the
// RMW happens at the L2 atomic units coherently across WGPs.
__device__ __forceinline__ void atomic_add_f32_hw(float* p, float v) {
    asm volatile("global_atomic_add_f32 %0, %1, off scope:SCOPE_DEV"
                 :: "v"(p), "v"(v) : "memory");
}

// ---------------- degree / normalization ----------------
__global__ void k_deg_init(float* __restrict__ deg, int n) {
    int i = blockIdx.x * blockDim.x + threadIdx.x;
    if (i < n) deg[i] = 1.0f;                      // self-loop contribution
}

__global__ void k_deg_count(const long long* __restrict__ dst, float* __restrict__ deg, int e) {
    int i = blockIdx.x * blockDim.x + threadIdx.x;
    if (i < e) atomic_add_f32_hw(&deg[(int)dst[i]], 1.0f);
}

__global__ void k_dinv(float* __restrict__ deg, int n) {
    int i = blockIdx.x * blockDim.x + threadIdx.x;
    if (i < n) deg[i] = rsqrtf(deg[i]);            // deg >= 1 always (self-loops)
}

// ---------------- WMMA f32 GEMM:  H[n x DO] = (relu?)X[n x 128] * W[128 x DO] ----------------
// One block = 16 rows. One wave per 16-column tile (NW = DO/16 waves).
// A-frag (16x4 f32, ISA §7.12.2): lane m=lane&15 -> row, kg=lane>>4 -> K pair {2kg, 2kg+1}.
// B-frag (4x16 f32): VGPR0 = row (k0+2kg), VGPR1 = row (k0+2kg+1), col n0+m.
// C/D (16x16 f32): acc[j] -> row 8*kg + j, col n0 + m.
template<int DO, bool RELU>
__global__ __launch_bounds__(32*(DO/16)) void k_gemm_wmma(
    const float* __restrict__ X, const float* __restrict__ W,
    float* __restrict__ H, int nrows)
{
    constexpr int NW = DO / 16;
    constexpr int SW = DO + 8;        // W LDS stride: 2*SW mod 64 == 16 -> half-waves hit disjoint banks
    constexpr int SX = DIN + 4;       // X LDS stride: banks = (4m + j) mod 64 -> conflict-free frag read
    __shared__ float sW[64 * SW];     // 64-row slab of W
    __shared__ float sX[16 * SX];     // 16 x 128 X tile

    const int tid  = threadIdx.x;
    const int row0 = blockIdx.x * 16;

    // stage X tile (apply ReLU for layer 2 here)
    for (int i = tid; i < 16 * DIN; i += 32 * NW) {
        int r = i >> 7, c = i & (DIN - 1);
        int gr = row0 + r; if (gr >= nrows) gr = nrows - 1;
        float v = X[(size_t)gr * DIN + c];
        if (RELU) v = fmaxf(v, 0.0f);
        sX[r * SX + c] = v;
    }

    const int lane = tid & 31;
    const int wv   = tid >> 5;        // column tile
    const int m    = lane & 15;
    const int kg   = lane >> 4;       // 0 or 1
    const int n0   = wv * 16;

    v8f acc = {};
    for (int kh = 0; kh < DIN; kh += 64) {
        __syncthreads();              // prior compute done / X tile staged
        for (int i = tid; i < 64 * DO; i += 32 * NW) {
            int r = i / DO, c = i % DO;
            sW[r * SW + c] = W[(size_t)(kh + r) * DO + c];
        }
        __syncthreads();
        #pragma unroll
        for (int k0 = 0; k0 < 64; k0 += 4) {
            const int ka = k0 + 2 * kg;
            v2f a, b;
            a.x = sX[m * SX + kh + ka];
            a.y = sX[m * SX + kh + ka + 1];
            b.x = sW[ka * SW + n0 + m];
            b.y = sW[(ka + 1) * SW + n0 + m];
            acc = __builtin_amdgcn_wmma_f32_16x16x4_f32(
                      false, a, false, b, (short)0, acc, false, false);
        }
    }

    #pragma unroll
    for (int j = 0; j < 8; ++j) {
        int gr = row0 + 8 * kg + j;
        if (gr < nrows) H[(size_t)gr * DO + n0 + m] = acc[j];
    }
}

// ---------------- self-loop + bias init:  O = G * dinv^2 + bias ----------------
template<int D>
__global__ void k_self_bias(const float* __restrict__ G, const float* __restrict__ dinv,
                            const float* __restrict__ bias, float* __restrict__ O, int n)
{
    int i = blockIdx.x * blockDim.x + threadIdx.x;
    if (i < n * D) {
        int v = i / D, f = i % D;
        float dv = dinv[v];
        O[i] = G[i] * dv * dv + bias[f];
    }
}

// ---------------- edge scatter: out[dst] += G[src] * dinv[src]*dinv[dst] ----------------
// One wave per edge; D/32 floats per lane; feature matrix is L2-resident (51MB < 192MB L2),
// so gathers + device-scope fp atomics run at L2 bandwidth, not HBM.
template<int D>
__global__ __launch_bounds__(256) void k_scatter(
    const float* __restrict__ G, const long long* __restrict__ src,
    const long long* __restrict__ dst, const float* __restrict__ dinv,
    float* __restrict__ out, int e)
{
    constexpr int VPL = D / 32;       // 4 (D=128) or 2 (D=64) floats per lane
    const int wid  = (int)((blockIdx.x * blockDim.x + threadIdx.x) >> 5);
    const int lane = threadIdx.x & 31;
    if (wid >= e) return;
    const int s = (int)src[wid];
    const int d = (int)dst[wid];
    const float nrm = dinv[s] * dinv[d];
    const float* gp = G   + (size_t)s * D + lane * VPL;
    float*       op = out + (size_t)d * D + lane * VPL;
    float v[VPL];
    #pragma unroll
    for (int j = 0; j < VPL; ++j) v[j] = gp[j];     // coalesces to b128/b64 gather
    #pragma unroll
    for (int j = 0; j < VPL; ++j) atomic_add_f32_hw(&op[j], v[j] * nrm);
}

extern "C" void kernel_launch(void* const* d_in, const int* in_sizes, int n_in,
                              void* d_out, int out_size, void* d_ws, size_t ws_size,
                              hipStream_t stream) {
    const float*     x   = (const float*)d_in[0];
    const long long* ei  = (const long long*)d_in[1];   // int64 [2, E]
    const float*     W1  = (const float*)d_in[2];
    const float*     b1  = (const float*)d_in[3];
    const float*     W2  = (const float*)d_in[4];
    const float*     b2  = (const float*)d_in[5];
    float*           out = (float*)d_out;

    const int n = in_sizes[0] / DIN;      // 100000 nodes
    const int e = in_sizes[1] / 2;        // 1600000 edges
    const long long* src = ei;
    const long long* dst = ei + e;

    // workspace carve-out
    float* dinv = (float*)d_ws;                   // n
    float* h    = dinv + n;                       // n*128 (reused for layer-2 product n*64)
    float* buf1 = h + (size_t)n * 128;            // n*128

    // normalization coefficients
    k_deg_init <<<(n + 255) / 256, 256, 0, stream>>>(dinv, n);
    k_deg_count<<<(e + 255) / 256, 256, 0, stream>>>(dst, dinv, e);
    k_dinv     <<<(n + 255) / 256, 256, 0, stream>>>(dinv, n);

    const int nblk = (n + 15) / 16;

    // ---- layer 1: buf1 = Ahat * (x @ W1) + b1   (relu deferred to layer-2 load) ----
    k_gemm_wmma<128, false><<<nblk, 256, 0, stream>>>(x, W1, h, n);
    k_self_bias<128><<<((size_t)n * 128 + 255) / 256, 256, 0, stream>>>(h, dinv, b1, buf1, n);
    k_scatter<128><<<(e + 7) / 8, 256, 0, stream>>>(h, src, dst, dinv, buf1, e);

    // ---- layer 2: out = Ahat * (relu(buf1) @ W2) + b2 ----
    float* g = h; // reuse: h is dead after layer-1 scatter
    k_gemm_wmma<64, true><<<nblk, 128, 0, stream>>>(buf1, W2, g, n);
    k_self_bias<64><<<((size_t)n * 64 + 255) / 256, 256, 0, stream>>>(g, dinv, b2, out, n);
    k_scatter<64><<<(e + 7) / 8, 256, 0, stream>>>(g, src, dst, dinv, out, e);
}